// SelectiveSSM_31361851195872
// MI455X (gfx1250) — compile-verified
//
#include <hip/hip_runtime.h>
#include <hip/hip_bf16.h>
#include <stdint.h>

typedef __attribute__((ext_vector_type(2))) float v2f;
typedef __attribute__((ext_vector_type(8))) float v8f;

#define D_MODEL   1024
#define STATE_DIM 16
#define DT_RANK   64
#define BATCH     4
#define SEQ_LEN   1024
#define BL        (BATCH * SEQ_LEN)   // 4096 rows
#define NCHUNK    8
#define CHUNK     128                 // SEQ_LEN / NCHUNK
#define LOG2E     1.4426950408889634f

// ---------------------------------------------------------------------------
// CDNA5 async global->LDS copy (ASYNCcnt-tracked, no VGPR round trip).
// ---------------------------------------------------------------------------
__device__ __forceinline__ void async_ld_b128(void* lds_dst, const void* gsrc) {
    asm volatile("global_load_async_to_lds_b128 %0, %1, off"
                 :: "v"((uint32_t)(uintptr_t)lds_dst), "v"(gsrc)
                 : "memory");
}
__device__ __forceinline__ void async_wait0() {
    asm volatile("s_wait_asynccnt 0x0" ::: "memory");
}

// ---------------------------------------------------------------------------
// Kernel 1: P = X @ W_delta (N=64), Bm = X @ W_B, Cm = X @ W_C
// 6 waves/block: waves 0-3 -> P tiles, wave 4 -> Bm, wave 5 -> Cm.
// X tile staged via async-to-LDS in two 16x512 halves, stride 516 dwords
// (516 % 64 == 4 -> conflict-free float2 A-frag reads).
// ---------------------------------------------------------------------------
__global__ __launch_bounds__(192) void ssm_proj_kernel(
    const float* __restrict__ X,
    const float* __restrict__ Wd,   // [1024,64]
    const float* __restrict__ Wb,   // [1024,16]
    const float* __restrict__ Wc,   // [1024,16]
    float* __restrict__ P,          // [4096,64]
    float* __restrict__ Bm,         // [4096,16]
    float* __restrict__ Cm)         // [4096,16]
{
    const int S = 516;
    __shared__ float ldsx[16 * 516];

    const int tid  = threadIdx.x;
    const int row0 = blockIdx.x * 16;
    const int lane = tid & 31;
    const int w    = tid >> 5;

    const int m     = lane & 15;          // A-row / B-col within tile
    const int khalf = (lane >> 4) << 1;   // 0 or 2
    const int rowhi = (lane >> 4) << 3;   // 0 or 8 (C/D layout)

    const float* Bsrc; int bstride, colbase; float* Out; int ostride;
    if (w < 4)       { Bsrc = Wd; bstride = 64; colbase = w * 16; Out = P;  ostride = 64; }
    else if (w == 4) { Bsrc = Wb; bstride = 16; colbase = 0;      Out = Bm; ostride = 16; }
    else             { Bsrc = Wc; bstride = 16; colbase = 0;      Out = Cm; ostride = 16; }

    v8f c = {};
    for (int half = 0; half < 2; ++half) {
        __syncthreads();
        for (int idx = tid; idx < 16 * 128; idx += 192) {
            int r  = idx >> 7;
            int c4 = (idx & 127) << 2;
            async_ld_b128(&ldsx[r * S + c4],
                          X + (size_t)(row0 + r) * D_MODEL + half * 512 + c4);
        }
        async_wait0();
        __syncthreads();

        #pragma unroll 4
        for (int k0 = 0; k0 < 512; k0 += 4) {
            const int kk = k0 + khalf;          // local k within half
            const int kg = half * 512 + kk;     // global k for weights
            v2f a;
            a.x = ldsx[m * S + kk];
            a.y = ldsx[m * S + kk + 1];
            v2f b;
            b.x = Bsrc[(size_t)kg       * bstride + colbase + m];
            b.y = Bsrc[(size_t)(kg + 1) * bstride + colbase + m];
            c = __builtin_amdgcn_wmma_f32_16x16x4_f32(
                    false, a, false, b, (short)0, c, false, false);
        }
    }

    #pragma unroll
    for (int r = 0; r < 8; ++r) {
        int mm = r + rowhi;
        Out[(size_t)(row0 + mm) * ostride + colbase + m] = c[r];
    }
}

// ---------------------------------------------------------------------------
// Kernel 2: delta = softplus(P @ W_dt + b_dt)   [4096,64] x [64,1024]
// ---------------------------------------------------------------------------
__global__ __launch_bounds__(256) void ssm_delta_kernel(
    const float* __restrict__ P,    // [4096,64]
    const float* __restrict__ Wdt,  // [64,1024]
    const float* __restrict__ bdt,  // [1024]
    float* __restrict__ Delta)      // [4096,1024]
{
    const int S = 68;
    __shared__ float ldsp[16 * 68];

    const int tid  = threadIdx.x;
    const int row0 = blockIdx.x * 16;
    {
        int r  = tid >> 4;
        int c4 = (tid & 15) << 2;
        async_ld_b128(&ldsp[r * S + c4],
                      P + (size_t)(row0 + r) * DT_RANK + c4);
    }
    async_wait0();
    __syncthreads();

    const int lane  = tid & 31;
    const int w     = tid >> 5;
    const int m     = lane & 15;
    const int khalf = (lane >> 4) << 1;
    const int rowhi = (lane >> 4) << 3;

    for (int nt = w; nt < 64; nt += 8) {
        const int col = nt * 16 + m;
        v8f c = {};
        #pragma unroll 4
        for (int k0 = 0; k0 < DT_RANK; k0 += 4) {
            const int kk = k0 + khalf;
            v2f a;
            a.x = ldsp[m * S + kk];
            a.y = ldsp[m * S + kk + 1];
            v2f b;
            b.x = Wdt[(size_t)kk       * D_MODEL + col];
            b.y = Wdt[(size_t)(kk + 1) * D_MODEL + col];
            c = __builtin_amdgcn_wmma_f32_16x16x4_f32(
                    false, a, false, b, (short)0, c, false, false);
        }
        const float bias = bdt[col];
        #pragma unroll
        for (int r = 0; r < 8; ++r) {
            float z  = c[r] + bias;
            float sp = (z > 20.0f) ? z : log1pf(__expf(z));   // softplus
            Delta[(size_t)(row0 + r + rowhi) * D_MODEL + col] = sp;
        }
    }
}

// ---------------------------------------------------------------------------
// Chunked scan, pass 1: per chunk, run recurrence from h=0. Emit chunk-local
// end state E[b,c,d,n] and chunk decay Pp[b,c,d,n] = exp2(a2[n]*sum(dt)).
// Grid: b (4) x chunk (8) x d-block (4) = 128 blocks, 256 threads.
// ---------------------------------------------------------------------------
__global__ __launch_bounds__(256) void ssm_scan_pass1(
    const float* __restrict__ X,      // [B,L,D]
    const float* __restrict__ A_log,  // [D,16]
    const float* __restrict__ Delta,  // [B,L,D]
    const float* __restrict__ Bm,     // [B,L,16]
    float* __restrict__ Ebuf,         // [B,C,D,16]
    float* __restrict__ Ppbuf)        // [B,C,D,16]
{
    __shared__ float ldsB[CHUNK * 16];

    const int tid  = threadIdx.x;
    const int bx   = blockIdx.x;
    const int dblk = bx & 3;
    const int cc   = (bx >> 2) & 7;
    const int b    = bx >> 5;
    const int d    = (dblk << 8) + tid;

    float a2[16], h[16];
    #pragma unroll
    for (int nn = 0; nn < 16; ++nn) {
        a2[nn] = -__expf(A_log[d * 16 + nn]) * LOG2E;   // A*log2(e)
        h[nn]  = 0.0f;
    }

    const size_t lbase = (size_t)b * SEQ_LEN + (size_t)cc * CHUNK;
    // stage B rows for this chunk (2048 contiguous floats)
    for (int q = tid; q < CHUNK * 16 / 4; q += 256) {
        async_ld_b128(&ldsB[q * 4], Bm + lbase * 16 + (size_t)q * 4);
    }
    async_wait0();
    __syncthreads();

    float sdt = 0.0f;
    for (int t = 0; t < CHUNK; ++t) {
        const size_t off = (lbase + t) * D_MODEL + d;
        const float dt  = Delta[off];
        const float xv  = X[off];
        const float dtx = dt * xv;
        sdt += dt;
        #pragma unroll
        for (int nn = 0; nn < 16; ++nn) {
            float dA = exp2f(dt * a2[nn]);
            h[nn] = dA * h[nn] + dtx * ldsB[t * 16 + nn];
        }
    }

    const size_t obase = (((size_t)b * NCHUNK + cc) * D_MODEL + d) * 16;
    #pragma unroll
    for (int nn = 0; nn < 16; ++nn) {
        Ebuf[obase + nn]  = h[nn];
        Ppbuf[obase + nn] = exp2f(sdt * a2[nn]);   // prod of decays, exact form
    }
}

// ---------------------------------------------------------------------------
// Chunked scan, pass 2: 8-step prefix over chunks per (b,d,n):
//   H[0]=0 ; H[c] = Pp[c-1]*H[c-1] + E[c-1]
// 65536 independent lanes.
// ---------------------------------------------------------------------------
__global__ __launch_bounds__(256) void ssm_scan_pass2(
    const float* __restrict__ Ebuf,   // [B,C,D,16]
    const float* __restrict__ Ppbuf,  // [B,C,D,16]
    float* __restrict__ Hbuf)         // [B,C,D,16]
{
    const int id = blockIdx.x * 256 + threadIdx.x;   // 0..65535
    const int b  = id >> 14;
    const int dn = id & 16383;                       // d*16 + n

    float h = 0.0f;
    #pragma unroll
    for (int cc = 0; cc < NCHUNK; ++cc) {
        const size_t idx = (((size_t)b * NCHUNK + cc) << 14) + dn;
        Hbuf[idx] = h;
        h = Ppbuf[idx] * h + Ebuf[idx];
    }
}

// ---------------------------------------------------------------------------
// Chunked scan, pass 3: re-run each chunk starting from the true initial
// state H[b,c,d,:], producing y. Same grid as pass 1.
// ---------------------------------------------------------------------------
__global__ __launch_bounds__(256) void ssm_scan_pass3(
    const float* __restrict__ X,      // [B,L,D]
    const float* __restrict__ A_log,  // [D,16]
    const float* __restrict__ Delta,  // [B,L,D]
    const float* __restrict__ Bm,     // [B,L,16]
    const float* __restrict__ Cm,     // [B,L,16]
    const float* __restrict__ Dvec,   // [D]
    const float* __restrict__ Hbuf,   // [B,C,D,16]
    float* __restrict__ Y)            // [B,L,D]
{
    __shared__ float ldsB[CHUNK * 16];
    __shared__ float ldsC[CHUNK * 16];

    const int tid  = threadIdx.x;
    const int bx   = blockIdx.x;
    const int dblk = bx & 3;
    const int cc   = (bx >> 2) & 7;
    const int b    = bx >> 5;
    const int d    = (dblk << 8) + tid;

    const size_t hbase = (((size_t)b * NCHUNK + cc) * D_MODEL + d) * 16;
    float a2[16], h[16];
    #pragma unroll
    for (int nn = 0; nn < 16; ++nn) {
        a2[nn] = -__expf(A_log[d * 16 + nn]) * LOG2E;
        h[nn]  = Hbuf[hbase + nn];
    }
    const float dd = Dvec[d];

    const size_t lbase = (size_t)b * SEQ_LEN + (size_t)cc * CHUNK;
    for (int q = tid; q < CHUNK * 16 / 4; q += 256) {
        async_ld_b128(&ldsB[q * 4], Bm + lbase * 16 + (size_t)q * 4);
        async_ld_b128(&ldsC[q * 4], Cm + lbase * 16 + (size_t)q * 4);
    }
    async_wait0();
    __syncthreads();

    for (int t = 0; t < CHUNK; ++t) {
        const size_t off = (lbase + t) * D_MODEL + d;
        const float dt  = Delta[off];
        const float xv  = X[off];
        const float dtx = dt * xv;
        float acc = 0.0f;
        #pragma unroll
        for (int nn = 0; nn < 16; ++nn) {
            float dA = exp2f(dt * a2[nn]);
            h[nn] = dA * h[nn] + dtx * ldsB[t * 16 + nn];
            acc  += h[nn] * ldsC[t * 16 + nn];
        }
        Y[off] = acc + dd * xv;
    }
}

// ---------------------------------------------------------------------------
// Fallback one-pass scan (used only if workspace is too small for chunking).
// ---------------------------------------------------------------------------
__global__ __launch_bounds__(256) void ssm_scan_simple(
    const float* __restrict__ X, const float* __restrict__ A_log,
    const float* __restrict__ Delta, const float* __restrict__ Bm,
    const float* __restrict__ Cm, const float* __restrict__ Dvec,
    float* __restrict__ Y)
{
    __shared__ float ldsB[64 * 16];
    __shared__ float ldsC[64 * 16];
    const int tid = threadIdx.x;
    const int b   = blockIdx.x >> 2;
    const int d   = ((blockIdx.x & 3) << 8) + tid;

    float a2[16], h[16];
    #pragma unroll
    for (int nn = 0; nn < 16; ++nn) {
        a2[nn] = -__expf(A_log[d * 16 + nn]) * LOG2E;
        h[nn]  = 0.0f;
    }
    const float dd = Dvec[d];
    const size_t rowbase = (size_t)b * SEQ_LEN;

    for (int l0 = 0; l0 < SEQ_LEN; l0 += 64) {
        __syncthreads();
        for (int idx = tid; idx < 64 * 16; idx += 256) {
            size_t g = (rowbase + l0 + (idx >> 4)) * 16 + (idx & 15);
            ldsB[idx] = Bm[g];
            ldsC[idx] = Cm[g];
        }
        __syncthreads();
        for (int i = 0; i < 64; ++i) {
            const size_t off = (rowbase + l0 + i) * D_MODEL + d;
            const float dt  = Delta[off];
            const float xv  = X[off];
            const float dtx = dt * xv;
            float acc = 0.0f;
            #pragma unroll
            for (int nn = 0; nn < 16; ++nn) {
                float dA = exp2f(dt * a2[nn]);
                h[nn] = dA * h[nn] + dtx * ldsB[i * 16 + nn];
                acc  += h[nn] * ldsC[i * 16 + nn];
            }
            Y[off] = acc + dd * xv;
        }
    }
}

// ---------------------------------------------------------------------------
extern "C" void kernel_launch(void* const* d_in, const int* in_sizes, int n_in,
                              void* d_out, int out_size, void* d_ws, size_t ws_size,
                              hipStream_t stream) {
    const float* x       = (const float*)d_in[0];
    const float* A_log   = (const float*)d_in[1];
    const float* W_delta = (const float*)d_in[2];
    const float* W_dt    = (const float*)d_in[3];
    const float* b_dt    = (const float*)d_in[4];
    const float* W_B     = (const float*)d_in[5];
    const float* W_C     = (const float*)d_in[6];
    const float* Dv      = (const float*)d_in[7];
    float* out = (float*)d_out;

    float* ws    = (float*)d_ws;
    float* P     = ws;                                   // [4096,64]
    float* Delta = P     + (size_t)BL * DT_RANK;         // [4096,1024]
    float* Bms   = Delta + (size_t)BL * D_MODEL;         // [4096,16]
    float* Cms   = Bms   + (size_t)BL * STATE_DIM;       // [4096,16]
    float* Ebuf  = Cms   + (size_t)BL * STATE_DIM;       // [B,C,D,16]
    float* Ppbuf = Ebuf  + (size_t)BATCH * NCHUNK * D_MODEL * STATE_DIM;
    float* Hbuf  = Ppbuf + (size_t)BATCH * NCHUNK * D_MODEL * STATE_DIM;

    const size_t need = ((size_t)(Hbuf - ws) +
                         (size_t)BATCH * NCHUNK * D_MODEL * STATE_DIM) * sizeof(float);

    ssm_proj_kernel<<<dim3(BL / 16), dim3(192), 0, stream>>>(
        x, W_delta, W_B, W_C, P, Bms, Cms);
    ssm_delta_kernel<<<dim3(BL / 16), dim3(256), 0, stream>>>(
        P, W_dt, b_dt, Delta);

    if (ws_size >= need) {
        ssm_scan_pass1<<<dim3(BATCH * NCHUNK * 4), dim3(256), 0, stream>>>(
            x, A_log, Delta, Bms, Ebuf, Ppbuf);
        ssm_scan_pass2<<<dim3(256), dim3(256), 0, stream>>>(
            Ebuf, Ppbuf, Hbuf);
        ssm_scan_pass3<<<dim3(BATCH * NCHUNK * 4), dim3(256), 0, stream>>>(
            x, A_log, Delta, Bms, Cms, Dv, Hbuf, out);
    } else {
        ssm_scan_simple<<<dim3(BATCH * 4), dim3(256), 0, stream>>>(
            x, A_log, Delta, Bms, Cms, Dv, out);
    }
}